// MU_BlockAttention_944892805632
// MI455X (gfx1250) — compile-verified
//
#include <hip/hip_runtime.h>
#include <hip/hip_bf16.h>

typedef __attribute__((ext_vector_type(2))) float v2f;
typedef __attribute__((ext_vector_type(8))) float v8f;

#define Bv 16
#define Tv 512
#define NT (Bv * Tv)          // 8192 tokens
#define RS2 0.70710678118654752f   // 1/sqrt(2)
#define RS8 0.35355339059327376f   // 1/sqrt(8)

__device__ __forceinline__ v8f wmma4(v2f a, v2f b, v8f c) {
  // D = A(16x4 f32) x B(4x16 f32) + C(16x16 f32)
  return __builtin_amdgcn_wmma_f32_16x16x4_f32(false, a, false, b, (short)0, c,
                                               false, false);
}

__device__ __forceinline__ float gelu_exact(float x) {
  return 0.5f * x * (1.0f + erff(x * RS2));
}

// ---- DPP-based wave reductions (VALU only; no LDS pipe / dscnt waits) ------
template <int CTRL>
__device__ __forceinline__ float dpp_mov(float v) {
  return __int_as_float(
      __builtin_amdgcn_mov_dpp(__float_as_int(v), CTRL, 0xF, 0xF, true));
}
// butterfly reduction within each 16-lane row
__device__ __forceinline__ float max16(float v) {
  v = fmaxf(v, dpp_mov<0xB1>(v));    // quad_perm(1,0,3,2)  == xor 1
  v = fmaxf(v, dpp_mov<0x4E>(v));    // quad_perm(2,3,0,1)  == xor 2
  v = fmaxf(v, dpp_mov<0x141>(v));   // row_half_mirror (combines quads in 8)
  v = fmaxf(v, dpp_mov<0x140>(v));   // row_mirror (combines 8-groups in 16)
  return v;
}
__device__ __forceinline__ float sum16(float v) {
  v += dpp_mov<0xB1>(v);
  v += dpp_mov<0x4E>(v);
  v += dpp_mov<0x141>(v);
  v += dpp_mov<0x140>(v);
  return v;
}
// exchange with the other 16-lane half (identity lane selects)
__device__ __forceinline__ float xhalf(float v) {
  return __int_as_float(__builtin_amdgcn_permlanex16(
      __float_as_int(v), __float_as_int(v), 0x76543210, 0xFEDCBA98, false,
      false));
}
__device__ __forceinline__ float sum32(float v) {
  v = sum16(v);
  return v + xhalf(v);
}
__device__ __forceinline__ float max32(float v) {
  v = max16(v);
  return fmaxf(v, xhalf(v));
}

// ---------------------------------------------------------------------------
// K1: extract 2x2 blocks + per-block QKV projection (tiny 4x12 matmuls).
// grid = NT, block = 192 (one thread per (block n, qkv component e)).
// q/k/v layout: [B, 32 heads, T, 2]   (head = block*2 + h)
// ---------------------------------------------------------------------------
__global__ void k_blk_qkv(const float* __restrict__ M,
                          const float* __restrict__ wqkv,
                          const float* __restrict__ bqkv,
                          float* __restrict__ qb, float* __restrict__ kb,
                          float* __restrict__ vb) {
  int tok = blockIdx.x;
  int j = threadIdx.x;              // 0..191
  int n = j / 12, e = j % 12;
  int b = tok / Tv, t = tok % Tv;
  int br = n >> 2, bc = n & 3;
  const float* Mr = M + (long)tok * 64;
  float x0 = Mr[(br * 2 + 0) * 8 + bc * 2 + 0];
  float x1 = Mr[(br * 2 + 0) * 8 + bc * 2 + 1];
  float x2 = Mr[(br * 2 + 1) * 8 + bc * 2 + 0];
  float x3 = Mr[(br * 2 + 1) * 8 + bc * 2 + 1];
  const float* w = wqkv + ((long)n * 12 + e) * 4;
  float v = x0 * w[0] + x1 * w[1] + x2 * w[2] + x3 * w[3] + bqkv[n * 12 + e];
  int c = e & 3;                    // component within q/k/v
  int h = c >> 1, d = c & 1;
  int head = n * 2 + h;
  long idx = (((long)b * 32 + head) * Tv + t) * 2 + d;
  if (e < 4)       qb[idx] = v;
  else if (e < 8)  kb[idx] = v;
  else             vb[idx] = v;
}

// ---------------------------------------------------------------------------
// K2: per-block attention, head_dim=2 -> VALU flash. One wave per (b,head,t).
// grid = B*32*T/8, block = 256 (8 waves). Output o: [B,T,16,4].
// ---------------------------------------------------------------------------
__global__ void k_blk_attn(const float* __restrict__ qb,
                           const float* __restrict__ kb,
                           const float* __restrict__ vb,
                           float* __restrict__ ob) {
  int lane = threadIdx.x & 31;
  long w = (long)blockIdx.x * 8 + (threadIdx.x >> 5);
  int t = (int)(w % Tv);
  int head = (int)((w / Tv) % 32);
  int b = (int)(w / ((long)Tv * 32));
  const float* qp = qb + (((long)b * 32 + head) * Tv + t) * 2;
  float q0 = qp[0] * RS2, q1 = qp[1] * RS2;
  const float* kp = kb + ((long)b * 32 + head) * Tv * 2;
  const float* vp = vb + ((long)b * 32 + head) * Tv * 2;
  float m = -3.4e38f, l = 0.f, o0 = 0.f, o1 = 0.f;
  for (int sc = 0; sc < Tv / 32; ++sc) {
    int s = sc * 32 + lane;
    float k0 = kp[s * 2 + 0], k1 = kp[s * 2 + 1];
    float scv = q0 * k0 + q1 * k1;
    float mn = fmaxf(m, max32(scv));
    float rs = __expf(m - mn);
    float p = __expf(scv - mn);
    l = l * rs + p;
    o0 = o0 * rs + p * vp[s * 2 + 0];
    o1 = o1 * rs + p * vp[s * 2 + 1];
    m = mn;
  }
  l = sum32(l);
  o0 = sum32(o0);
  o1 = sum32(o1);
  if (lane == 0) {
    float inv = 1.f / l;
    int n = head >> 1, hh = head & 1;
    float* op = ob + (((long)b * Tv + t) * 16 + n) * 4 + hh * 2;
    op[0] = o0 * inv;
    op[1] = o1 * inv;
  }
}

// ---------------------------------------------------------------------------
// K3: per-block 4x4 output projection -> all_blocks [B,T,64].
// grid = NT, block = 64.
// ---------------------------------------------------------------------------
__global__ void k_blk_out(const float* __restrict__ ob,
                          const float* __restrict__ wout,
                          const float* __restrict__ bout,
                          float* __restrict__ ab) {
  long tok = blockIdx.x;
  int j = threadIdx.x, n = j >> 2, e = j & 3;
  const float* o = ob + (tok * 16 + n) * 4;
  const float* w = wout + ((long)n * 4 + e) * 4;
  ab[tok * 64 + j] =
      o[0] * w[0] + o[1] * w[1] + o[2] * w[2] + o[3] * w[3] + bout[n * 4 + e];
}

// ---------------------------------------------------------------------------
// Generic f32 WMMA GEMM tile kernel: out[rows,N] = A[rows,K] @ W[N,K]^T + bias
// MODE 0: plain   MODE 1: GELU epilogue   MODE 2: + residual
// One wave per 16x16 output tile; grid = (rows/16, N/16), block = 32.
// ---------------------------------------------------------------------------
template <int K, int N, int MODE>
__global__ void k_gemm(const float* __restrict__ A, const float* __restrict__ W,
                       const float* __restrict__ bias,
                       const float* __restrict__ resid,
                       float* __restrict__ out) {
  int lane = threadIdx.x & 31;
  int half = lane >> 4;
  int lm = lane & 15;
  int mt = blockIdx.x, nt = blockIdx.y;
  const float* arow = A + (long)(mt * 16 + lm) * K;
  const float* wrow = W + (long)(nt * 16 + lm) * K;
  v8f acc = {};
#pragma unroll
  for (int k0 = 0; k0 < K; k0 += 4) {
    int ka = k0 + half * 2;
    v2f a = {arow[ka], arow[ka + 1]};     // A fragment 16x4
    v2f bm = {wrow[ka], wrow[ka + 1]};    // B fragment: B[k][n] = W[n][k]
    acc = wmma4(a, bm, acc);
  }
  int col = nt * 16 + lm;
  float bvl = bias[col];
#pragma unroll
  for (int r = 0; r < 8; ++r) {
    long orow = (long)mt * 16 + half * 8 + r;
    float v = acc[r] + bvl;
    if (MODE == 1) v = gelu_exact(v);
    if (MODE == 2) v += resid[orow * N + col];
    out[orow * N + col] = v;
  }
}

// ---------------------------------------------------------------------------
// K5: cross-block MHA, flash style with f32 WMMA (K=8 scores, K=16 P@V via
// LDS re-striping). grid = B*T/16, block = 256 (wave h owns head h).
// cqkv rows: [cq(0..63) | ck(64..127) | cv(128..191)]. Saves per-row (m,l).
// Softmax row reductions are 16-lane DPP butterflies (match C-layout rows).
// ---------------------------------------------------------------------------
__global__ void k_cross_attn(const float* __restrict__ cqkv,
                             float* __restrict__ co, float* __restrict__ mrow,
                             float* __restrict__ lrow) {
  __shared__ float plds[8 * 256];
  int lane = threadIdx.x & 31;
  int h = threadIdx.x >> 5;
  int b = blockIdx.x / (Tv / 16);
  int qt = blockIdx.x % (Tv / 16);
  int t0 = qt * 16;
  int lm = lane & 15;
  int half = lane >> 4;
  int koff = half * 2;
  const float* base = cqkv + (long)b * Tv * 192;
  float* pl = plds + h * 256;

  v2f aq[2];
#pragma unroll
  for (int c = 0; c < 2; ++c) {
    const float* qp = base + (long)(t0 + lm) * 192 + h * 8 + c * 4 + koff;
    aq[c].x = qp[0] * RS8;
    aq[c].y = qp[1] * RS8;
  }
  float m[8], l[8];
#pragma unroll
  for (int r = 0; r < 8; ++r) { m[r] = -3.4e38f; l[r] = 0.f; }
  v8f o = {};

  for (int st = 0; st < Tv / 16; ++st) {
    int s0 = st * 16;
    // scores S = Q K^T (two K=4 WMMAs)
    v8f sc = {};
#pragma unroll
    for (int c = 0; c < 2; ++c) {
      const float* kp = base + (long)(s0 + lm) * 192 + 64 + h * 8 + c * 4 + koff;
      v2f bm = {kp[0], kp[1]};
      sc = wmma4(aq[c], bm, sc);
    }
    // online softmax per row (row = half*8 + r, spread over 16-lane group)
#pragma unroll
    for (int r = 0; r < 8; ++r) {
      float mn = fmaxf(m[r], max16(sc[r]));
      float rs = __expf(m[r] - mn);
      float p = __expf(sc[r] - mn);
      l[r] = l[r] * rs + sum16(p);
      o[r] *= rs;
      m[r] = mn;
      pl[(half * 8 + r) * 16 + lm] = p;  // C-layout -> LDS (wave-private)
    }
    // O += P @ V  (four K=4 WMMAs; P re-striped to A-layout via LDS)
#pragma unroll
    for (int c = 0; c < 4; ++c) {
      v2f ap = {pl[lm * 16 + c * 4 + koff], pl[lm * 16 + c * 4 + koff + 1]};
      const float* vp =
          base + (long)(s0 + c * 4 + koff) * 192 + 128 + h * 8 + (lm & 7);
      v2f bvv = {vp[0], vp[192]};
      o = wmma4(ap, bvv, o);
    }
  }
#pragma unroll
  for (int r = 0; r < 8; ++r) {
    float inv = 1.f / l[r];
    int orow = t0 + half * 8 + r;
    if (lm < 8) co[((long)b * Tv + orow) * 64 + h * 8 + lm] = o[r] * inv;
    if (lm == 0) {
      mrow[((long)b * 8 + h) * Tv + orow] = m[r];
      lrow[((long)b * 8 + h) * Tv + orow] = l[r];
    }
  }
}

// ---------------------------------------------------------------------------
// K6: entropy of head-averaged attention. Recompute scores streaming using the
// saved flash (m,l) stats. One wave per (b,t) row; grid = NT/8, block = 256.
// ---------------------------------------------------------------------------
__global__ void k_entropy(const float* __restrict__ cqkv,
                          const float* __restrict__ mrow,
                          const float* __restrict__ lrow,
                          float* __restrict__ ent) {
  __shared__ float qs[8][64];
  int lane = threadIdx.x & 31;
  int wid = threadIdx.x >> 5;
  long row = (long)blockIdx.x * 8 + wid;  // b*T + t
  int b = (int)(row / Tv), t = (int)(row % Tv);
  const float* base = cqkv + (long)b * Tv * 192;
  qs[wid][lane] = base[(long)t * 192 + lane] * RS8;
  qs[wid][lane + 32] = base[(long)t * 192 + lane + 32] * RS8;
  float mh[8], lh[8];
#pragma unroll
  for (int h = 0; h < 8; ++h) {
    mh[h] = mrow[((long)b * 8 + h) * Tv + t];
    lh[h] = 1.f / lrow[((long)b * 8 + h) * Tv + t];
  }
  float acc = 0.f;
  for (int scix = 0; scix < Tv / 32; ++scix) {
    int s = scix * 32 + lane;
    const float* kp = base + (long)s * 192 + 64;
    float a = 0.f;
#pragma unroll
    for (int h = 0; h < 8; ++h) {
      float d = 0.f;
#pragma unroll
      for (int k = 0; k < 8; ++k) d += qs[wid][h * 8 + k] * kp[h * 8 + k];
      a += __expf(d - mh[h]) * lh[h];
    }
    a *= 0.125f;
    acc -= a * __logf(a + 1e-9f);
  }
  acc = sum32(acc);
  if (lane == 0) ent[row] = acc;
}

// ---------------------------------------------------------------------------
// LayerNorm over 64: one wave per row, 2 elems per lane. grid=NT/8, block=256.
// ---------------------------------------------------------------------------
__global__ void k_ln(const float* __restrict__ x, const float* __restrict__ g,
                     const float* __restrict__ bb, float* __restrict__ out) {
  int lane = threadIdx.x & 31;
  long row = (long)blockIdx.x * 8 + (threadIdx.x >> 5);
  const float* xr = x + row * 64;
  float a0 = xr[lane * 2], a1 = xr[lane * 2 + 1];
  float s = sum32(a0 + a1);
  float sq = sum32(a0 * a0 + a1 * a1);
  float mu = s * (1.f / 64.f);
  float var = sq * (1.f / 64.f) - mu * mu;
  float r = rsqrtf(var + 1e-5f);
  float* orow = out + row * 64;
  orow[lane * 2] = (a0 - mu) * r * g[lane * 2] + bb[lane * 2];
  orow[lane * 2 + 1] = (a1 - mu) * r * g[lane * 2 + 1] + bb[lane * 2 + 1];
}

// ---------------------------------------------------------------------------
// K12: sensitivity MLP (17->32->16) + sigmoid gate + final update.
// One thread per token. grid = 32, block = 256.
// ---------------------------------------------------------------------------
__global__ void k_sens_gate(const float* __restrict__ x2,
                            const float* __restrict__ Min,
                            const int* __restrict__ tok_ids,
                            const float* __restrict__ aff,
                            const float* __restrict__ ent,
                            const float* __restrict__ sw1,
                            const float* __restrict__ sb1,
                            const float* __restrict__ sw2,
                            const float* __restrict__ sb2,
                            const float* __restrict__ sbase,
                            float* __restrict__ out) {
  long tok = (long)blockIdx.x * 256 + threadIdx.x;
  int id = tok_ids[tok];
  float feat[17];
#pragma unroll
  for (int j = 0; j < 16; ++j) feat[j] = aff[(long)id * 16 + j];
  feat[16] = ent[tok];
  float hbuf[32];
#pragma unroll
  for (int i = 0; i < 32; ++i) {
    float s = sb1[i];
#pragma unroll
    for (int j = 0; j < 17; ++j) s += sw1[i * 17 + j] * feat[j];
    hbuf[i] = gelu_exact(s);
  }
  const float* xr = x2 + tok * 64;
  const float* mr = Min + tok * 64;
  float* orow = out + tok * 64;
#pragma unroll
  for (int n = 0; n < 16; ++n) {
    float s = sb2[n];
#pragma unroll
    for (int j = 0; j < 32; ++j) s += sw2[n * 32 + j] * hbuf[j];
    float sens = (1.f / (1.f + __expf(-s))) * sbase[n];
#pragma unroll
    for (int d = 0; d < 4; ++d) {
      int i = n * 4 + d;
      orow[i] = mr[i] + (xr[i] - mr[i]) * sens;
    }
  }
}

// ---------------------------------------------------------------------------
extern "C" void kernel_launch(void* const* d_in, const int* in_sizes, int n_in,
                              void* d_out, int out_size, void* d_ws,
                              size_t ws_size, hipStream_t stream) {
  const float* M = (const float*)d_in[0];
  const int* token_ids = (const int*)d_in[1];
  const float* blk_w_qkv = (const float*)d_in[2];
  const float* blk_b_qkv = (const float*)d_in[3];
  const float* blk_w_out = (const float*)d_in[4];
  const float* blk_b_out = (const float*)d_in[5];
  const float* cross_w_qkv = (const float*)d_in[6];
  const float* cross_b_qkv = (const float*)d_in[7];
  const float* cross_w_out = (const float*)d_in[8];
  const float* cross_b_out = (const float*)d_in[9];
  const float* ln1_g = (const float*)d_in[10];
  const float* ln1_b = (const float*)d_in[11];
  const float* ffn_w1 = (const float*)d_in[12];
  const float* ffn_b1 = (const float*)d_in[13];
  const float* ffn_w2 = (const float*)d_in[14];
  const float* ffn_b2 = (const float*)d_in[15];
  const float* ln2_g = (const float*)d_in[16];
  const float* ln2_b = (const float*)d_in[17];
  const float* sens_base = (const float*)d_in[18];
  const float* tok_affinity = (const float*)d_in[19];
  const float* sens_w1 = (const float*)d_in[20];
  const float* sens_b1 = (const float*)d_in[21];
  const float* sens_w2 = (const float*)d_in[22];
  const float* sens_b2 = (const float*)d_in[23];
  float* out = (float*)d_out;

  float* ws = (float*)d_ws;
  float* qb = ws;                      // [B,32,T,2]  524288
  float* kb = qb + 524288;
  float* vb = kb + 524288;
  float* ob = vb + 524288;             // [NT,64]
  float* ab = ob + 524288;             // all_blocks [NT,64]
  float* cqkv = ab + 524288;           // [NT,192]
  float* co = cqkv + (long)NT * 192;   // [NT,64]
  float* mrow = co + 524288;           // [B,8,T]
  float* lrow = mrow + 65536;
  float* ent = lrow + 65536;           // [NT]
  float* x0 = ent + NT;                // pre-LN1
  float* x1 = x0 + 524288;             // post-LN1
  float* hf = x1 + 524288;             // FFN hidden [NT,256]
  float* y = hf + (long)NT * 256;      // pre-LN2
  float* x2 = y + 524288;              // post-LN2

  // 1. block QKV projections
  k_blk_qkv<<<NT, 192, 0, stream>>>(M, blk_w_qkv, blk_b_qkv, qb, kb, vb);
  // 2. 16 tiny per-block attentions (VALU flash, head_dim=2)
  k_blk_attn<<<(Bv * 32 * Tv) / 8, 256, 0, stream>>>(qb, kb, vb, ob);
  // 3. per-block output projection -> all_blocks
  k_blk_out<<<NT, 64, 0, stream>>>(ob, blk_w_out, blk_b_out, ab);
  // 4. cross QKV GEMM [NT,64]@[64,192]  (WMMA)
  k_gemm<64, 192, 0><<<dim3(NT / 16, 192 / 16), 32, 0, stream>>>(
      ab, cross_w_qkv, cross_b_qkv, nullptr, cqkv);
  // 5. cross attention flash (WMMA), saves (m,l) per row
  k_cross_attn<<<(Bv * Tv) / 16, 256, 0, stream>>>(cqkv, co, mrow, lrow);
  // 6. head-averaged attention entropy
  k_entropy<<<NT / 8, 256, 0, stream>>>(cqkv, mrow, lrow, ent);
  // 7. cross out-proj GEMM + residual (WMMA)
  k_gemm<64, 64, 2><<<dim3(NT / 16, 64 / 16), 32, 0, stream>>>(
      co, cross_w_out, cross_b_out, ab, x0);
  // 8. LayerNorm 1
  k_ln<<<NT / 8, 256, 0, stream>>>(x0, ln1_g, ln1_b, x1);
  // 9. FFN up GEMM + exact GELU (WMMA)
  k_gemm<64, 256, 1><<<dim3(NT / 16, 256 / 16), 32, 0, stream>>>(
      x1, ffn_w1, ffn_b1, nullptr, hf);
  // 10. FFN down GEMM + residual (WMMA)
  k_gemm<256, 64, 2><<<dim3(NT / 16, 64 / 16), 32, 0, stream>>>(
      hf, ffn_w2, ffn_b2, x1, y);
  // 11. LayerNorm 2
  k_ln<<<NT / 8, 256, 0, stream>>>(y, ln2_g, ln2_b, x2);
  // 12. sensitivity MLP + gated update
  k_sens_gate<<<NT / 256, 256, 0, stream>>>(x2, M, token_ids, tok_affinity, ent,
                                            sens_w1, sens_b1, sens_w2, sens_b2,
                                            sens_base, out);
  (void)in_sizes; (void)n_in; (void)out_size; (void)ws_size;
}